// GraphEncoder_88871463289211
// MI455X (gfx1250) — compile-verified
//
#include <hip/hip_runtime.h>

// ---------------------------------------------------------------------------
// GCN encoder for MI455X (gfx1250, wave32).
//   deg -> dinv -> norm                        (elementwise + f32 atomics)
//   h1 = x @ W1        via V_WMMA_F32_16X16X4_F32 (fp32; compute is free here,
//                       the scatter stage is the 23.3 TB/s roofline limiter)
//   agg1 = scatter_add(h1[src]*norm -> dst)    (bandwidth-dominant stage)
//   BN+ReLU (training-mode batch stats; GCN bias cancels under BN)
//   h2 = act1 @ W2, scatter, BN+ReLU -> d_out
// A tiles are staged to LDS by the Tensor Data Mover (tensor_load_to_lds),
// using TDM's pad feature to insert the 4-dword/row bank-conflict padding.
// ---------------------------------------------------------------------------

typedef __attribute__((ext_vector_type(2))) float v2f;
typedef __attribute__((ext_vector_type(8))) float v8f;
typedef __attribute__((ext_vector_type(4))) unsigned int v4u;
typedef __attribute__((ext_vector_type(4))) int v4i;
typedef __attribute__((ext_vector_type(8))) int v8i;

#define KDIM 256          // inner dim of both GEMMs
#define TM   32           // rows per block tile
#define LDSP (KDIM + 4)   // padded LDS row stride (floats) -> conflict-free

// ---------------------------------------------------------------------------
// C[M x NCOLS] = A[M x KDIM] * B[KDIM x NCOLS], all row-major fp32.
// 8 waves / block; each wave owns 16x16 output tiles; K-loop of 16x16x4 WMMAs
// on two interleaved accumulators to break the WMMA->WMMA RAW chain.
// ---------------------------------------------------------------------------
template <int NCOLS>
__global__ __launch_bounds__(256) void gemm_wmma_f32(
    const float* __restrict__ A, const float* __restrict__ B,
    float* __restrict__ C, int M) {
  __shared__ float As[TM * LDSP];

  const int row0 = blockIdx.x * TM;
  const int tid  = threadIdx.x;

  // --- TDM: DMA the TM x KDIM f32 tile into LDS with 4-dword row padding ---
  if (tid < 32) {  // one wave issues the descriptor (TDM ignores EXEC)
    const unsigned long long ga =
        (unsigned long long)(const void*)(A + (size_t)row0 * KDIM);
    const unsigned int lds = (unsigned int)(uintptr_t)&As[0];
    const unsigned int rows = (unsigned int)(M - row0);  // OOB rows zero-fill

    v4u g0;
    g0[0] = 1u;                                        // count=1 (valid, user)
    g0[1] = lds;                                       // D#.lds_addr
    g0[2] = (unsigned int)(ga & 0xFFFFFFFFu);          // global_addr[31:0]
    g0[3] = (unsigned int)((ga >> 32) & 0x01FFFFFFu) | (2u << 30);  // type=2

    v8i g1;
    // data_size=4B(2), pad_enable, pad_interval=256dw(7), pad_amount=4dw(3)
    g1[0] = (int)((2u << 16) | (1u << 20) | (7u << 22) | (3u << 25));
    g1[1] = (int)(((unsigned)KDIM & 0xFFFFu) << 16);   // tensor_dim0 lo16
    g1[2] = (int)((((unsigned)KDIM >> 16) & 0xFFFFu) | ((rows & 0xFFFFu) << 16));
    g1[3] = (int)(((rows >> 16) & 0xFFFFu) | ((unsigned)KDIM << 16)); // tile_dim0
    g1[4] = (int)TM;                                   // tile_dim1=32, dim2=0
    g1[5] = (int)KDIM;                                 // tensor_dim0_stride lo
    g1[6] = 0;
    g1[7] = 0;

    v4i z4 = {0, 0, 0, 0};
#if __has_include(<hip/amd_detail/amd_gfx1250_TDM.h>)
    v8i z8 = {0, 0, 0, 0, 0, 0, 0, 0};
    __builtin_amdgcn_tensor_load_to_lds(g0, g1, z4, z4, z8, 0);
#else
    __builtin_amdgcn_tensor_load_to_lds(g0, g1, z4, z4, 0);
#endif
    __builtin_amdgcn_s_wait_tensorcnt(0);
  }
  __syncthreads();

  const int wave = tid >> 5;
  const int lane = tid & 31;
  const int half = lane >> 4;   // 0: K pair {0,1}, 1: K pair {2,3}
  const int l16  = lane & 15;

  constexpr int NT16   = NCOLS / 16;
  constexpr int ntiles = (TM / 16) * NT16;
  const bool full = (row0 + TM) <= M;   // block-uniform store guard

  for (int t = wave; t < ntiles; t += 8) {
    const int m0 = (t / NT16) * 16;
    const int n0 = (t % NT16) * 16;
    const float* Bp = B + n0 + l16 + (size_t)(2 * half) * NCOLS;
    const float* Ap = As + (m0 + l16) * LDSP + 2 * half;

    v8f acc0 = {}, acc1 = {};
#pragma unroll 4
    for (int k = 0; k < KDIM; k += 8) {
      // A frag (16x4 f32): lanes 0-15 K={k,k+1}, lanes 16-31 K={k+2,k+3}
      v2f a0 = *(const v2f*)(Ap + k);
      v2f a1 = *(const v2f*)(Ap + k + 4);
      v2f b0, b1;
      b0.x = Bp[(size_t)(k + 0) * NCOLS];
      b0.y = Bp[(size_t)(k + 1) * NCOLS];
      b1.x = Bp[(size_t)(k + 4) * NCOLS];
      b1.y = Bp[(size_t)(k + 5) * NCOLS];
      acc0 = __builtin_amdgcn_wmma_f32_16x16x4_f32(
          false, a0, false, b0, (short)0, acc0, false, false);
      acc1 = __builtin_amdgcn_wmma_f32_16x16x4_f32(
          false, a1, false, b1, (short)0, acc1, false, false);
    }
    v8f acc = acc0 + acc1;

    // D layout: VGPR r -> (M = m0 + half*8 + r, N = n0 + l16)
    const int rb = row0 + m0 + half * 8;
    float* Cp = C + (size_t)rb * NCOLS + n0 + l16;
    if (full) {
#pragma unroll
      for (int r = 0; r < 8; ++r) Cp[(size_t)r * NCOLS] = acc[r];
    } else {
#pragma unroll
      for (int r = 0; r < 8; ++r)
        if (rb + r < M) Cp[(size_t)r * NCOLS] = acc[r];
    }
  }
}

// ---------------------------------------------------------------------------
// Degree / normalization
// ---------------------------------------------------------------------------
__global__ void deg_kernel(const int* __restrict__ dst, float* __restrict__ deg,
                           int E) {
  const int e = blockIdx.x * blockDim.x + threadIdx.x;
  if (e < E) unsafeAtomicAdd(&deg[dst[e]], 1.0f);
}

__global__ void dinv_kernel(float* __restrict__ deg, int n) {
  const int i = blockIdx.x * blockDim.x + threadIdx.x;
  if (i < n) {
    const float d = deg[i];
    deg[i] = d > 0.f ? __frsqrt_rn(fmaxf(d, 1.f)) : 0.f;
  }
}

__global__ void norm_kernel(const int* __restrict__ src,
                            const int* __restrict__ dst,
                            const float* __restrict__ dinv,
                            float* __restrict__ norm, int E) {
  const int e = blockIdx.x * blockDim.x + threadIdx.x;
  if (e < E) norm[e] = dinv[src[e]] * dinv[dst[e]];
}

// ---------------------------------------------------------------------------
// out[dst[e], f] += h[src[e], f] * norm[e]   (the bandwidth-bound stage)
// 256 threads = 256/F edge "slots"; prefetch next gathered row (gfx1250
// global_prefetch_b8) since the gather address is data-dependent.
// ---------------------------------------------------------------------------
template <int F>
__global__ __launch_bounds__(256) void scatter_add(
    const float* __restrict__ h, const float* __restrict__ norm,
    const int* __restrict__ src, const int* __restrict__ dst,
    float* __restrict__ out, int E, int iters) {
  constexpr int SLOTS = 256 / F;
  const int slot = threadIdx.x / F;
  const int f    = threadIdx.x % F;
  long e = (long)blockIdx.x * SLOTS * iters + slot;
  for (int it = 0; it < iters; ++it, e += SLOTS) {
    if (e < E) {
      const long en = e + SLOTS;
      if (en < E)
        __builtin_prefetch(&h[(size_t)src[en] * F + f], 0, 1);
      const float v = h[(size_t)src[e] * F + f] * norm[e];
      unsafeAtomicAdd(&out[(size_t)dst[e] * F + f], v);
    }
  }
}

// ---------------------------------------------------------------------------
// BatchNorm (training-mode batch statistics) + ReLU, two-pass
// ---------------------------------------------------------------------------
template <int F>
__global__ void bn_stats(const float* __restrict__ x, float* __restrict__ sum,
                         float* __restrict__ sumsq, int n, int npb) {
  const int f  = threadIdx.x;               // blockDim.x == F
  const int n0 = blockIdx.x * npb;
  int nend = n0 + npb;
  if (nend > n) nend = n;
  float s = 0.f, sq = 0.f;
  for (int i = n0; i < nend; ++i) {
    const float v = x[(size_t)i * F + f];
    s += v;
    sq += v * v;
  }
  unsafeAtomicAdd(&sum[f], s);
  unsafeAtomicAdd(&sumsq[f], sq);
}

__global__ void bn_finalize(const float* __restrict__ sum,
                            const float* __restrict__ sumsq,
                            const float* __restrict__ gamma,
                            const float* __restrict__ beta,
                            float* __restrict__ scale,
                            float* __restrict__ shift, int F, float invN) {
  const int f = threadIdx.x;
  if (f < F) {
    const float mean = sum[f] * invN;
    const float var  = fmaxf(sumsq[f] * invN - mean * mean, 0.f);
    const float sc   = gamma[f] * __frsqrt_rn(var + 1e-5f);
    scale[f] = sc;
    shift[f] = beta[f] - mean * sc;   // GCN bias cancels under BN mean-subtract
  }
}

template <int F>
__global__ __launch_bounds__(256) void bn_apply_relu(
    const float* __restrict__ x, const float* __restrict__ scale,
    const float* __restrict__ shift, float* __restrict__ y, size_t total) {
  const size_t i = (size_t)blockIdx.x * 256 + threadIdx.x;
  if (i < total) {
    const int f = (int)(i & (size_t)(F - 1));
    const float v = fmaf(x[i], scale[f], shift[f]);
    y[i] = v > 0.f ? v : 0.f;
  }
}

// ---------------------------------------------------------------------------
extern "C" void kernel_launch(void* const* d_in, const int* in_sizes, int n_in,
                              void* d_out, int out_size, void* d_ws,
                              size_t ws_size, hipStream_t stream) {
  const float* x   = (const float*)d_in[0];
  const int*   ei  = (const int*)d_in[1];
  const float* W1  = (const float*)d_in[2];
  const float* g1  = (const float*)d_in[4];
  const float* bt1 = (const float*)d_in[5];
  const float* W2  = (const float*)d_in[6];
  const float* g2  = (const float*)d_in[8];
  const float* bt2 = (const float*)d_in[9];
  float* out = (float*)d_out;

  const int Nn  = in_sizes[0] / 256;   // 100000 nodes
  const int E   = in_sizes[1] / 2;     // 3.2M edges
  const int LAT = in_sizes[6] / 256;   // 64

  const int* src = ei;
  const int* dst = ei + E;

  // Workspace layout (~220 MB): deg | norm | stats(1024) | bufH | bufA
  float* ws   = (float*)d_ws;
  float* deg  = ws;                       // N        (becomes dinv in place)
  float* norm = deg + (size_t)Nn;         // E
  float* sum  = norm + (size_t)E;         // 256
  float* ssq  = sum + 256;                // 256
  float* scl  = ssq + 256;                // 256
  float* shf  = scl + 256;                // 256
  float* bufH = shf + 256;                // N*256 (h1; later h2 | agg2)
  float* bufA = bufH + (size_t)Nn * 256;  // N*256 (agg1 -> act1 in place)
  float* h2   = bufH;                     // N*64
  float* agg2 = bufH + (size_t)Nn * LAT;  // N*64

  // ---- normalization coefficients -----------------------------------------
  hipMemsetAsync(deg, 0, (size_t)Nn * 4, stream);
  deg_kernel<<<(E + 255) / 256, 256, 0, stream>>>(dst, deg, E);
  dinv_kernel<<<(Nn + 255) / 256, 256, 0, stream>>>(deg, Nn);
  norm_kernel<<<(E + 255) / 256, 256, 0, stream>>>(src, dst, deg, norm, E);

  // ---- layer 1: GEMM -> scatter -> BN+ReLU --------------------------------
  const int gblocks = (Nn + TM - 1) / TM;
  gemm_wmma_f32<256><<<gblocks, 256, 0, stream>>>(x, W1, bufH, Nn);

  hipMemsetAsync(bufA, 0, (size_t)Nn * 256 * 4, stream);
  {
    const int iters = 8, per_blk = (256 / 256) * iters;
    scatter_add<256><<<(E + per_blk - 1) / per_blk, 256, 0, stream>>>(
        bufH, norm, src, dst, bufA, E, iters);
  }

  hipMemsetAsync(sum, 0, 512 * 4, stream);
  bn_stats<256><<<(Nn + 255) / 256, 256, 0, stream>>>(bufA, sum, ssq, Nn, 256);
  bn_finalize<<<1, 256, 0, stream>>>(sum, ssq, g1, bt1, scl, shf, 256,
                                     1.0f / (float)Nn);
  {
    const size_t tot = (size_t)Nn * 256;
    bn_apply_relu<256><<<(unsigned)((tot + 255) / 256), 256, 0, stream>>>(
        bufA, scl, shf, bufA, tot);
  }

  // ---- layer 2: GEMM -> scatter -> BN+ReLU --------------------------------
  gemm_wmma_f32<64><<<gblocks, 256, 0, stream>>>(bufA, W2, h2, Nn);

  hipMemsetAsync(agg2, 0, (size_t)Nn * LAT * 4, stream);
  {
    const int iters = 8, per_blk = (256 / 64) * iters;
    scatter_add<64><<<(E + per_blk - 1) / per_blk, 256, 0, stream>>>(
        h2, norm, src, dst, agg2, E, iters);
  }

  hipMemsetAsync(sum, 0, 512 * 4, stream);
  bn_stats<64><<<(Nn + 255) / 256, 64, 0, stream>>>(agg2, sum, ssq, Nn, 256);
  bn_finalize<<<1, 64, 0, stream>>>(sum, ssq, g2, bt2, scl, shf, 64,
                                    1.0f / (float)Nn);
  {
    const size_t tot = (size_t)Nn * LAT;
    bn_apply_relu<64><<<(unsigned)((tot + 255) / 256), 256, 0, stream>>>(
        agg2, scl, shf, out, tot);
  }
}